// Quixer_90237262889135
// MI455X (gfx1250) — compile-verified
//
#include <hip/hip_runtime.h>
#include <stdint.h>

#define VOCAB   50257
#define EMB     512
#define BATCH   16
#define NWORDS  16
#define DIM     256
#define NROTS   64
#define DEG     3
#define NTILES  ((VOCAB + 15) / 16)   // 3142
#define HSTRIDE 516                   // 512 + 4 pad -> conflict-free LDS banks

typedef __attribute__((ext_vector_type(2))) float v2f;
typedef __attribute__((ext_vector_type(8))) float v8f;

// ---------------- quantum gate helpers (state resident in LDS) ----------------
// dim-256 state, 128 threads; each gate partitions indices into disjoint pairs,
// so only an inter-gate barrier is needed.

__device__ __forceinline__ void apply_ry(float2* st, int tid, int p, float th) {
    float s, c;
    __sincosf(th * 0.5f, &s, &c);
    int mask = (1 << p) - 1;
    int i0 = ((tid & ~mask) << 1) | (tid & mask);
    int i1 = i0 | (1 << p);
    float2 a0 = st[i0], a1 = st[i1];
    st[i0] = make_float2(c * a0.x - s * a1.x, c * a0.y - s * a1.y);
    st[i1] = make_float2(s * a0.x + c * a1.x, s * a0.y + c * a1.y);
    __syncthreads();
}

// controlled-RX: control bit pc must be 1; RX = [[c,-is],[-is,c]]
__device__ __forceinline__ void apply_crx(float2* st, int tid, int pc, int pt, float th) {
    float s, c;
    __sincosf(th * 0.5f, &s, &c);
    if (tid < 64) {
        int blo = pc < pt ? pc : pt;
        int bhi = pc < pt ? pt : pc;
        int x = tid;
        x = ((x >> blo) << (blo + 1)) | (x & ((1 << blo) - 1));
        x = ((x >> bhi) << (bhi + 1)) | (x & ((1 << bhi) - 1));
        int ib = x | (1 << pc);
        int i0 = ib;
        int i1 = ib | (1 << pt);
        float2 a0 = st[i0], a1 = st[i1];
        st[i0] = make_float2(c * a0.x + s * a1.y, c * a0.y - s * a1.x);
        st[i1] = make_float2(s * a0.y + c * a1.x, c * a1.y - s * a0.x);
    }
    __syncthreads();
}

// circuit 14; qubit w <-> bit position (7 - w)
__device__ void sim14(float2* st, int tid, const float* prm, int layers) {
    int idx = 0;
    for (int L = 0; L < layers; ++L) {
        for (int i = 0; i < 8; ++i)  apply_ry (st, tid, 7 - i, prm[idx++]);
        for (int i = 7; i >= 0; --i) apply_crx(st, tid, 7 - i, 7 - ((i + 1) & 7), prm[idx++]);
        for (int i = 0; i < 8; ++i)  apply_ry (st, tid, 7 - i, prm[idx++]);
        apply_crx(st, tid, 0, 1, prm[idx++]);                       // c=7, t=6
        for (int i = 0; i < 7; ++i)  apply_crx(st, tid, 7 - i, 7 - ((i + 7) & 7), prm[idx++]);
    }
}

// ---------------- kernels ----------------

// base state, acc = poly[0]*base, normalized complex mix coefficients
__global__ void k_init(const float* poly, const float* mixri,
                       float2* work, float2* acc, float2* coeffs) {
    __shared__ float mag[16];
    int b = blockIdx.x, i = threadIdx.x;            // 16 x 256
    float v = (i == 0) ? 1.f : 0.f;
    work[b * DIM + i] = make_float2(v, 0.f);
    acc [b * DIM + i] = make_float2(poly[0] * v, 0.f);
    if (b == 0) {
        if (i < 16) {
            float re = mixri[2 * i], im = mixri[2 * i + 1];
            mag[i] = sqrtf(re * re + im * im);
        }
        __syncthreads();
        if (i < 16) {
            float tot = 0.f;
            for (int w = 0; w < 16; ++w) tot += mag[w];
            tot = fmaxf(tot, 1e-12f);
            coeffs[i] = make_float2(mixri[2 * i] / tot, mixri[2 * i + 1] / tot);
        }
    }
}

// wparams = emb @ emb2rot_W^T + b  via V_WMMA_F32_16X16X4_F32
// M = 256 (b*16+w), N = 64, K = 512. 64 waves, one 16x16 tile each.
__global__ void __launch_bounds__(256) k_wparams(const int* x, const float* embW,
                                                 const float* e2rW, const float* e2rb,
                                                 float* wparams) {
    int lane = threadIdx.x & 31;
    int wave = (blockIdx.x * blockDim.x + threadIdx.x) >> 5;   // 0..63
    int mt = wave >> 2;                                        // 0..15
    int nt = wave & 3;                                         // 0..3
    int col = lane & 15;
    int klo = (lane >> 4) << 1;                                // 0 or 2
    int tok = x[mt * 16 + col];                                // flat (b,w) row
    const float* arow = embW + (long)tok * EMB;
    const float* brow = e2rW + (nt * 16 + col) * EMB;
    v8f acc = {0.f, 0.f, 0.f, 0.f, 0.f, 0.f, 0.f, 0.f};
    #pragma unroll 4
    for (int k = 0; k < EMB; k += 4) {
        v2f a = *(const v2f*)(arow + k + klo);
        v2f b = *(const v2f*)(brow + k + klo);
        acc = __builtin_amdgcn_wmma_f32_16x16x4_f32(false, a, false, b,
                                                    (short)0, acc, false, false);
    }
    int n = nt * 16 + col;
    float bias = e2rb[n];
    int mbase = mt * 16 + ((lane >> 4) << 3);                  // lanes>=16 hold M+8
    #pragma unroll
    for (int r = 0; r < 8; ++r)
        wparams[(mbase + r) * NROTS + n] = acc[r] + bias;
}

// one block per (b,w) state: replicate work[b], run 2-layer circuit
__global__ void __launch_bounds__(128) k_sim(const float2* work, const float* wparams,
                                             float2* st_all) {
    __shared__ float2 st[DIM];
    __shared__ float  prm[NROTS];
    int s = blockIdx.x;                 // 0..255
    int b = s >> 4;
    int tid = threadIdx.x;              // 128
    st[tid]       = work[b * DIM + tid];
    st[tid + 128] = work[b * DIM + tid + 128];
    if (tid < NROTS) prm[tid] = wparams[s * NROTS + tid];
    __syncthreads();
    sim14(st, tid, prm, 2);
    st_all[s * DIM + tid]       = st[tid];
    st_all[s * DIM + tid + 128] = st[tid + 128];
}

// work[b][i] = sum_w coeffs[w]*st[b,w,i];  acc += poly[d]*work
__global__ void k_contract(const float2* st_all, const float2* coeffs, const float* poly,
                           int d, float2* work, float2* acc) {
    int b = blockIdx.x, i = threadIdx.x;          // 16 x 256
    float2 sum = make_float2(0.f, 0.f);
    #pragma unroll
    for (int w = 0; w < NWORDS; ++w) {
        float2 c = coeffs[w];
        float2 a = st_all[(b * NWORDS + w) * DIM + i];
        sum.x += c.x * a.x - c.y * a.y;
        sum.y += c.x * a.y + c.y * a.x;
    }
    work[b * DIM + i] = sum;
    float pd = poly[d];
    float2 ac = acc[b * DIM + i];
    ac.x += pd * sum.x;
    ac.y += pd * sum.y;
    acc[b * DIM + i] = ac;
}

// normalize, final_probs, 1-layer circuit, X/Y/Z measurements
__global__ void __launch_bounds__(128) k_final(const float2* acc, const float* poly,
                                               const float* qff, float* exps, float* fps) {
    __shared__ float2 st[DIM];
    __shared__ float  red[3][128];
    int b = blockIdx.x, tid = threadIdx.x;        // 16 x 128
    float psum = fabsf(poly[0]) + fabsf(poly[1]) + fabsf(poly[2]) + fabsf(poly[3]);
    float inv = 1.f / psum;
    float2 m0 = acc[b * DIM + tid];
    float2 m1 = acc[b * DIM + tid + 128];
    m0.x *= inv; m0.y *= inv; m1.x *= inv; m1.y *= inv;
    red[0][tid] = m0.x * m0.x + m0.y * m0.y + m1.x * m1.x + m1.y * m1.y;
    __syncthreads();
    for (int o = 64; o > 0; o >>= 1) {
        if (tid < o) red[0][tid] += red[0][tid + o];
        __syncthreads();
    }
    float nrm = sqrtf(red[0][0]);
    float invn = 1.f / fmaxf(nrm, 1e-12f);
    st[tid]       = make_float2(m0.x * invn, m0.y * invn);
    st[tid + 128] = make_float2(m1.x * invn, m1.y * invn);
    __syncthreads();
    sim14(st, tid, qff, 1);                       // uses qff_params[0..31]
    for (int w = 0; w < 8; ++w) {
        int p = 7 - w;
        int mask = (1 << p) - 1;
        int i0 = ((tid & ~mask) << 1) | (tid & mask);
        int i1 = i0 | (1 << p);
        float2 a0 = st[i0], a1 = st[i1];
        red[0][tid] = a0.x * a1.x + a0.y * a1.y;                      // Re(conj(a0)*a1)
        red[1][tid] = a0.x * a1.y - a0.y * a1.x;                      // Im
        red[2][tid] = (a0.x * a0.x + a0.y * a0.y) - (a1.x * a1.x + a1.y * a1.y);
        __syncthreads();
        for (int o = 64; o > 0; o >>= 1) {
            if (tid < o) {
                red[0][tid] += red[0][tid + o];
                red[1][tid] += red[1][tid + o];
                red[2][tid] += red[2][tid + o];
            }
            __syncthreads();
        }
        if (tid == 0) {
            exps[b * 24 + w]      = 2.f * red[0][0];
            exps[b * 24 + 8 + w]  = 2.f * red[1][0];
            exps[b * 24 + 16 + w] = red[2][0];
        }
        __syncthreads();
    }
    if (tid == 0) fps[b] = nrm;
}

// h = relu(exps @ ff1_W^T + ff1_b)   (16 x 24) * (24 x 512)
__global__ void k_ff1(const float* exps, const float* ff1W, const float* ff1b, float* h) {
    int gid = blockIdx.x * blockDim.x + threadIdx.x;   // 8192
    int b = gid >> 9;
    int e = gid & 511;
    const float* ex = exps + b * 24;
    const float* wr = ff1W + e * 24;
    float s = ff1b[e];
    #pragma unroll
    for (int m = 0; m < 24; ++m) s += ex[m] * wr[m];
    h[gid] = fmaxf(s, 0.f);
}

__global__ void k_mean(const float* fps, float* out_scalar) {
    if (threadIdx.x == 0) {
        float s = 0.f;
        for (int i = 0; i < BATCH; ++i) s += fps[i];
        out_scalar[0] = s * (1.f / BATCH);
    }
}

// op = h @ ff2_W^T + ff2_b  -> 16 x 50257. Memory-bound: streams 103 MB of ff2_W
// exactly once -> non-temporal loads/stores so the stream doesn't churn L2.
// h staged into padded LDS rows via the gfx1250 async-to-LDS DMA path
// (ASYNCcnt-tracked, bypasses VGPRs). One wave per 16-column vocab tile, f32 WMMA.
__global__ void __launch_bounds__(256) k_gemm2(const float* h, const float* ff2W,
                                               const float* ff2b, float* op) {
    __shared__ float hs[BATCH * HSTRIDE];
    int tid = threadIdx.x;
    // 16 rows x 128 16-byte chunks; each chunk stays inside one row so the
    // padded LDS row stride (HSTRIDE) is preserved.
    for (int chunk = tid; chunk < BATCH * (EMB / 4); chunk += 256) {
        int row = chunk >> 7;
        int c4  = chunk & 127;
        unsigned lds_addr = (unsigned)(uintptr_t)&hs[row * HSTRIDE + c4 * 4];
        unsigned long long gaddr =
            (unsigned long long)(uintptr_t)(h + row * EMB + c4 * 4);
        asm volatile("global_load_async_to_lds_b128 %0, %1, off"
                     :: "v"(lds_addr), "v"(gaddr) : "memory");
    }
    asm volatile("s_wait_asynccnt 0x0" ::: "memory");
    __syncthreads();

    int wave = tid >> 5;
    int lane = tid & 31;
    int tile = blockIdx.x * 8 + wave;
    if (tile >= NTILES) return;
    int nbase = tile * 16;
    int col = lane & 15;
    int klo = (lane >> 4) << 1;                       // 0 or 2
    int v = nbase + col;
    int vc = v < VOCAB ? v : (VOCAB - 1);             // clamp loads; store guarded
    const float* brow = ff2W + (long)vc * EMB;
    const float* arow = hs + col * HSTRIDE;           // A row M == lane&15
    v8f acc = {0.f, 0.f, 0.f, 0.f, 0.f, 0.f, 0.f, 0.f};
    #pragma unroll 8
    for (int k = 0; k < EMB; k += 4) {
        v2f a = *(const v2f*)(arow + k + klo);
        v2f bb = __builtin_nontemporal_load((const v2f*)(brow + k + klo));
        acc = __builtin_amdgcn_wmma_f32_16x16x4_f32(false, a, false, bb,
                                                    (short)0, acc, false, false);
    }
    if (v < VOCAB) {
        float bias = ff2b[v];
        int mbase = (lane >> 4) << 3;                 // lanes>=16 hold rows 8..15
        #pragma unroll
        for (int r = 0; r < 8; ++r) {
            float val = acc[r] + bias;
            __builtin_nontemporal_store(val, &op[(long)(mbase + r) * VOCAB + v]);
        }
    }
}

// ---------------- launcher ----------------
extern "C" void kernel_launch(void* const* d_in, const int* in_sizes, int n_in,
                              void* d_out, int out_size, void* d_ws, size_t ws_size,
                              hipStream_t stream) {
    (void)in_sizes; (void)n_in; (void)out_size; (void)ws_size;
    const int*   x     = (const int*)d_in[0];
    const float* embW  = (const float*)d_in[1];
    const float* e2rW  = (const float*)d_in[2];
    const float* e2rb  = (const float*)d_in[3];
    const float* poly  = (const float*)d_in[4];
    const float* mixri = (const float*)d_in[5];
    const float* qff   = (const float*)d_in[6];
    const float* ff1W  = (const float*)d_in[7];
    const float* ff1b  = (const float*)d_in[8];
    const float* ff2W  = (const float*)d_in[9];
    const float* ff2b  = (const float*)d_in[10];
    float* out = (float*)d_out;

    float* ws = (float*)d_ws;
    size_t off = 0;
    float*  wparams = ws + off;            off += 256 * NROTS;        // 16384
    float2* coeffs  = (float2*)(ws + off); off += NWORDS * 2;         // 32
    float2* work    = (float2*)(ws + off); off += BATCH * DIM * 2;    // 8192
    float2* acc     = (float2*)(ws + off); off += BATCH * DIM * 2;    // 8192
    float2* st_all  = (float2*)(ws + off); off += 256 * DIM * 2;      // 131072
    float*  exps    = ws + off;            off += BATCH * 24;         // 384
    float*  fps     = ws + off;            off += BATCH;              // 16
    float*  h       = ws + off;            off += BATCH * EMB;        // 8192

    k_init<<<16, 256, 0, stream>>>(poly, mixri, work, acc, coeffs);
    k_wparams<<<8, 256, 0, stream>>>(x, embW, e2rW, e2rb, wparams);
    for (int d = 1; d <= DEG; ++d) {
        k_sim<<<256, 128, 0, stream>>>(work, wparams, st_all);
        k_contract<<<16, 256, 0, stream>>>(st_all, coeffs, poly, d, work, acc);
    }
    k_final<<<16, 128, 0, stream>>>(acc, poly, qff, exps, fps);
    k_ff1<<<32, 256, 0, stream>>>(exps, ff1W, ff1b, h);
    k_mean<<<1, 32, 0, stream>>>(fps, out + (size_t)BATCH * VOCAB);
    k_gemm2<<<(NTILES + 7) / 8, 256, 0, stream>>>(h, ff2W, ff2b, out);
}